// EdgeConv_71116068488000
// MI455X (gfx1250) — compile-verified
//
#include <hip/hip_runtime.h>

typedef __attribute__((ext_vector_type(2))) float v2f;
typedef __attribute__((ext_vector_type(8))) float v8f;

#define NE 16384
#define NF 128
#define NUNITS 128
#define NNODES 4096
#define BROW 256          // dwords per pair-row in LDS B buffer (no padding; XOR swizzle)

// ---------------- kernel 0: zero the node-aggregation buffer ----------------
__global__ void zero_agg_kernel(float* agg) {
    int i = blockIdx.x * blockDim.x + threadIdx.x;   // NNODES*NF threads
    agg[i] = 0.0f;
}

// ---------------- kernel 1: segment_sum(edge_feature, edge_dst) -------------
__global__ void scatter_agg_kernel(const float* __restrict__ feat,
                                   const int* __restrict__ dst,
                                   float* __restrict__ agg) {
    int idx = blockIdx.x * blockDim.x + threadIdx.x; // NE*NF threads
    int e = idx >> 7;
    int f = idx & (NF - 1);
    atomicAdd(&agg[(size_t)dst[e] * NF + f], feat[idx]);
}

// ---------------- kernel 2: message = agg[src] - duplicate-pair sum ---------
// keys (24-bit src*4096+dst) live in LDS; one wave per edge scans all keys
// with a ballot. Matches are rare (~1 self hit), so this is a cheap LDS scan,
// not the reference's O(E^2) GEMM.
__global__ void message_kernel(const float* __restrict__ feat,
                               const int* __restrict__ src,
                               const int* __restrict__ dst,
                               const float* __restrict__ agg,
                               float* __restrict__ message) {
    __shared__ unsigned int keys[NE];                // 64 KB LDS
    const int tid = threadIdx.x;
    for (int i = tid; i < NE; i += blockDim.x)
        keys[i] = ((unsigned)src[i] << 12) | (unsigned)dst[i];
    __syncthreads();

    const int wave = tid >> 5;
    const int lane = tid & 31;

    for (int t = 0; t < 16; ++t) {
        const int e = blockIdx.x * 128 + wave * 16 + t;
        const unsigned mykey = keys[e];
        float r0 = 0.f, r1 = 0.f, r2 = 0.f, r3 = 0.f;
        for (int base = 0; base < NE; base += 32) {
            const unsigned k = keys[base + lane];
            unsigned long long m = __ballot(k == mykey);   // wave32: low 32 bits
            while (m) {
                const int j = base + (__ffsll((long long)m) - 1);
                m &= m - 1;
                const float* fj = feat + (size_t)j * NF;
                r0 += fj[lane];
                r1 += fj[lane + 32];
                r2 += fj[lane + 64];
                r3 += fj[lane + 96];
            }
        }
        const float* ag = agg + (size_t)(mykey >> 12) * NF;
        float* mrow = message + (size_t)e * NF;
        mrow[lane]      = ag[lane]      - r0;
        mrow[lane + 32] = ag[lane + 32] - r1;
        mrow[lane + 64] = ag[lane + 64] - r2;
        mrow[lane + 96] = ag[lane + 96] - r3;
    }
}

// ---------------- kernel 3: out = relu(concat(feat,message) @ W + b) --------
// Each wave computes a 16x128 output strip with V_WMMA_F32_16X16X4_F32
// (8 C tiles, 64 K-steps). B is double-buffered through LDS in 32 KB K=64
// chunks stored pair-interleaved ((k,n),(k+1,n) adjacent) so each fragment is
// one ds_load_b64 into an even VGPR pair, with an XOR bank swizzle
// (col ^= (pair_row&1)<<4) so the two lane-halves hit complementary banks.
__global__ void __launch_bounds__(256)
gemm_kernel(const float* __restrict__ feat,
            const float* __restrict__ message,
            const float* __restrict__ kern,   // [256,128] row-major
            const float* __restrict__ bias,   // [128]
            float* __restrict__ out) {        // [NE,128]
    __shared__ float Bs[2][32 * BROW];         // 2 x 32 KB

    const int tid  = threadIdx.x;
    const int wave = tid >> 5;
    const int lane = tid & 31;
    const int hl   = lane >> 4;                // 0 = lanes 0-15, 1 = lanes 16-31
    const int l16  = lane & 15;

    const int row0 = blockIdx.x * 128 + wave * 16;   // this wave's 16 rows
    const int row  = row0 + l16;

    // staging registers for one 64x128 B chunk: this thread owns 4 groups,
    // group g = tid + 256*j -> pair-row kp = g>>5 (k = 2kp, 2kp+1),
    // float4 column nq = g&31 (n = 4*nq .. 4*nq+3)
    float4 se[4], so[4];

    auto load_chunk = [&](int c) {
        const float4* srcv = (const float4*)(kern + (size_t)c * 64 * NUNITS);
#pragma unroll
        for (int j = 0; j < 4; ++j) {
            const int g  = tid + 256 * j;
            const int kp = g >> 5;
            const int nq = g & 31;
            se[j] = srcv[(2 * kp)     * (NUNITS / 4) + nq];
            so[j] = srcv[(2 * kp + 1) * (NUNITS / 4) + nq];
        }
    };
    auto store_chunk = [&](int buf) {
#pragma unroll
        for (int j = 0; j < 4; ++j) {
            const int g  = tid + 256 * j;
            const int kp = g >> 5;
            const int nq = g & 31;
            const int n0 = (nq * 4) ^ ((kp & 1) << 4);   // XOR bank swizzle
            float* base = &Bs[buf][kp * BROW + n0 * 2];
            v2f t0; t0.x = se[j].x; t0.y = so[j].x;
            v2f t1; t1.x = se[j].y; t1.y = so[j].y;
            v2f t2; t2.x = se[j].z; t2.y = so[j].z;
            v2f t3; t3.x = se[j].w; t3.y = so[j].w;
            ((v2f*)base)[0] = t0;
            ((v2f*)base)[1] = t1;
            ((v2f*)base)[2] = t2;
            ((v2f*)base)[3] = t3;
        }
    };

    v8f acc[8];
#pragma unroll
    for (int i = 0; i < 8; ++i) acc[i] = (v8f){};

    load_chunk(0);
    store_chunk(0);
    __syncthreads();

    for (int c = 0; c < 4; ++c) {              // 4 chunks of K=64
        const int buf = c & 1;
        if (c < 3) load_chunk(c + 1);          // overlap next fetch with WMMA

        // A rows come from edge_feature (k<128) or message (k>=128)
        const float* Abase =
            (c < 2 ? feat : message) + (size_t)row * NF + (c & 1) * 64;

#pragma unroll
        for (int ks = 0; ks < 16; ++ks) {      // K=4 per WMMA step
            const int klo = ks * 4 + (hl << 1);        // even
            v2f a = *(const v2f*)(Abase + klo);        // A 16x4 frag (b64)
            const float* brow = &Bs[buf][(ks * 2 + hl) * BROW];
#pragma unroll
            for (int nt = 0; nt < 8; ++nt) {
                const int ncol = (nt * 16 + l16) ^ (hl << 4);  // un-swizzle
                v2f b = *(const v2f*)(brow + ncol * 2);        // one ds_load_b64
                acc[nt] = __builtin_amdgcn_wmma_f32_16x16x4_f32(
                    false, a, false, b, (short)0, acc[nt], false, false);
            }
        }

        if (c < 3) {
            store_chunk(buf ^ 1);              // prior contents already consumed
            __syncthreads();
        }
    }

    // epilogue: bias + relu + store (C/D layout: VGPR r -> rows r / r+8)
#pragma unroll
    for (int nt = 0; nt < 8; ++nt) {
        const int n  = nt * 16 + l16;
        const float bv = bias[n];
#pragma unroll
        for (int r = 0; r < 8; ++r) {
            const int m = r + (hl << 3);
            const float v = acc[nt][r] + bv;
            out[(size_t)(row0 + m) * NUNITS + n] = fmaxf(v, 0.0f);
        }
    }
}

// ---------------------------------------------------------------------------
extern "C" void kernel_launch(void* const* d_in, const int* in_sizes, int n_in,
                              void* d_out, int out_size, void* d_ws, size_t ws_size,
                              hipStream_t stream) {
    const float* feat = (const float*)d_in[0];   // [NE,NF]
    const int*   src  = (const int*)d_in[1];     // [NE]
    const int*   dst  = (const int*)d_in[2];     // [NE]
    const float* kern = (const float*)d_in[3];   // [256,128]
    const float* bias = (const float*)d_in[4];   // [128]
    float*       out  = (float*)d_out;           // [NE,128]

    // workspace layout: agg [NNODES*NF] (2 MB) | message [NE*NF] (8 MB)
    float* agg     = (float*)d_ws;
    float* message = agg + (size_t)NNODES * NF;

    zero_agg_kernel<<<(NNODES * NF) / 256, 256, 0, stream>>>(agg);
    scatter_agg_kernel<<<(NE * NF) / 256, 256, 0, stream>>>(feat, dst, agg);
    message_kernel<<<NE / 128, 256, 0, stream>>>(feat, src, dst, agg, message);
    gemm_kernel<<<NE / 128, 256, 0, stream>>>(feat, message, kern, bias, out);
}